// GLNModule_36378372997706
// MI455X (gfx1250) — compile-verified
//
#include <hip/hip_runtime.h>
#include <hip/hip_bf16.h>
#include <math.h>

typedef float v2f __attribute__((ext_vector_type(2)));
typedef float v8f __attribute__((ext_vector_type(8)));

#define EPSV 0.001f

__device__ __forceinline__ float sigmoid_clipf(float y) {
    float p = 1.0f / (1.0f + expf(-y));
    return fminf(fmaxf(p, EPSV), 1.0f - EPSV);
}
__device__ __forceinline__ float logitf_(float p) {
    return logf(p) - log1pf(-p);
}

// ---------------------------------------------------------------------------
// Kernel 1: gating indices + first-layer logit input.
// Block l in {0,1,2}: g_l = pack(side @ N_l > O_l), x_l[0] = logit(B_l).
// Block 0 also writes x0[1+i] = clip(side[i], eps, 1-eps)
//   (== logit(sigmoid(clip(side)))).
// ---------------------------------------------------------------------------
__global__ void gln_setup(const float* __restrict__ side,
                          const float* __restrict__ N0, const float* __restrict__ N1,
                          const float* __restrict__ N2,
                          const float* __restrict__ O0, const float* __restrict__ O1,
                          const float* __restrict__ O2,
                          const float* __restrict__ B0, const float* __restrict__ B1,
                          const float* __restrict__ B2,
                          float* __restrict__ x0, float* __restrict__ x1,
                          float* __restrict__ x2, int* __restrict__ gout)
{
    const int l = blockIdx.x;
    const float* Nl = (l == 0) ? N0 : ((l == 1) ? N1 : N2);
    const float* Ol = (l == 0) ? O0 : ((l == 1) ? O1 : O2);
    const float* Bl = (l == 0) ? B0 : ((l == 1) ? B1 : B2);
    float* xl = (l == 0) ? x0 : ((l == 1) ? x1 : x2);
    const int t = threadIdx.x;

    float acc[8];
#pragma unroll
    for (int j = 0; j < 8; ++j) acc[j] = 0.0f;

    for (int i = t; i < 1024; i += 256) {
        const float s = side[i];
#pragma unroll
        for (int j = 0; j < 8; ++j) acc[j] += s * Nl[i * 8 + j];
    }

    __shared__ float red[8][256];
#pragma unroll
    for (int j = 0; j < 8; ++j) red[j][t] = acc[j];
    __syncthreads();
    for (int off = 128; off > 0; off >>= 1) {
        if (t < off) {
#pragma unroll
            for (int j = 0; j < 8; ++j) red[j][t] += red[j][t + off];
        }
        __syncthreads();
    }

    if (t == 0) {
        int g = 0;
#pragma unroll
        for (int j = 0; j < 8; ++j)
            if (red[j][0] > Ol[j]) g |= 1 << (7 - j);   // POWERS = [2^7 .. 2^0]
        gout[l] = g;
        xl[0] = logitf_(Bl[0]);                          // bias channel
    }

    if (l == 0) {
        for (int i = t; i < 1024; i += 256)
            x0[1 + i] = fminf(fmaxf(side[i], EPSV), 1.0f - EPSV);
    }
}

// ---------------------------------------------------------------------------
// Kernel 2: gated GEMV via V_WMMA_F32_16X16X4_F32.
//   y[n] = sum_k x[k] * W[g][k*N + n],  k in [0,K), n in [0,N)
// One wave per 16-column tile. A-fragment carries x in matrix row M=0 only
// (lane 0: K=0,1 ; lane 16: K=2,3 per the CDNA5 16x4 f32 A layout); B-fragment
// carries two W rows per VGPR-pair with the same lane-half/K pairing.
// D row M=0 lives in acc[0], lanes 0..15 (N = lane).
// Main K loop is check-free (K rounded down to a multiple of 4); one
// wave-uniform predicated tail handles the remainder so EXEC is all-ones for
// every WMMA.
// final_mode=0: out[n] = logit(clip(sigmoid(y)))   (next layer's input)
// final_mode=1: out[n] = clip(sigmoid(y))
// ---------------------------------------------------------------------------
__global__ void gln_gemv_wmma(const float* __restrict__ x,  // [K] logit-space input
                              const float* __restrict__ W,  // [256, K, N]
                              const int* __restrict__ gidx, int gslot,
                              float* __restrict__ out,      // [N]
                              int K, int N, int final_mode)
{
    __shared__ float sx[1056];
    const int tid  = threadIdx.x;
    const int lane = tid & 31;
    const int wave = tid >> 5;

    const int g = gidx[gslot];
    const float* __restrict__ Wg = W + (size_t)g * (size_t)K * (size_t)N;

    const int Kpad = (K + 3) & ~3;
    for (int i = tid; i < Kpad; i += blockDim.x) sx[i] = (i < K) ? x[i] : 0.0f;
    __syncthreads();

    const int ntile = (blockIdx.x * (blockDim.x >> 5) + wave) * 16;
    if (ntile >= N) return;                    // wave-uniform exit

    const int  col   = ntile + (lane & 15);
    const int  khalf = (lane >> 4) << 1;       // 0 for lanes 0-15, 2 for lanes 16-31
    const bool arow0 = (lane & 15) == 0;       // lanes 0 and 16 carry A row M=0

    // Per-lane streaming pointer: row (khalf) of this k-step, column `col`.
    const float* __restrict__ wp = Wg + (size_t)khalf * N + col;
    const size_t rowN  = (size_t)N;            // +1 k-step within pair
    const size_t stepN = (size_t)4 * N;        // +4 k per iteration

    v8f acc = {0.f, 0.f, 0.f, 0.f, 0.f, 0.f, 0.f, 0.f};

    const int Kmain = K & ~3;
#pragma unroll 4
    for (int k = 0; k < Kmain; k += 4) {
        v2f a, b;
        a.x = arow0 ? sx[k + khalf]     : 0.0f;
        a.y = arow0 ? sx[k + khalf + 1] : 0.0f;
        b.x = __builtin_nontemporal_load(wp);          // W[k+khalf][col]
        b.y = __builtin_nontemporal_load(wp + rowN);   // W[k+khalf+1][col]
        wp += stepN;
        acc = __builtin_amdgcn_wmma_f32_16x16x4_f32(
            /*neg_a=*/false, a, /*neg_b=*/false, b,
            /*c_mod=*/(short)0, acc, /*reuse_a=*/false, /*reuse_b=*/false);
    }

    if (Kmain < K) {                           // wave-uniform tail (K % 4 != 0)
        const int kr = Kmain + khalf;
        v2f a, b;
        a.x = arow0 ? sx[kr]     : 0.0f;       // sx zero-padded past K
        a.y = arow0 ? sx[kr + 1] : 0.0f;
        b.x = (kr     < K) ? __builtin_nontemporal_load(wp)        : 0.0f;
        b.y = (kr + 1 < K) ? __builtin_nontemporal_load(wp + rowN) : 0.0f;
        acc = __builtin_amdgcn_wmma_f32_16x16x4_f32(
            false, a, false, b, (short)0, acc, false, false);
    }

    if (lane < 16 && col < N) {
        const float p = sigmoid_clipf(acc[0]);           // D row M=0 -> VGPR0, N = lane
        out[col] = final_mode ? p : logitf_(p);
    }
}

// ---------------------------------------------------------------------------
// Kernel 3: final 513-length dot (output dim 1) + sigmoid/clip -> d_out[0].
// ---------------------------------------------------------------------------
__global__ void gln_final(const float* __restrict__ x2, const float* __restrict__ W2,
                          const int* __restrict__ gidx, float* __restrict__ out)
{
    const int g = gidx[2];
    const float* __restrict__ w = W2 + (size_t)g * 513;
    __shared__ float red[256];
    float acc = 0.0f;
    for (int i = threadIdx.x; i < 513; i += 256) acc += x2[i] * w[i];
    red[threadIdx.x] = acc;
    __syncthreads();
    for (int off = 128; off > 0; off >>= 1) {
        if (threadIdx.x < off) red[threadIdx.x] += red[threadIdx.x + off];
        __syncthreads();
    }
    if (threadIdx.x == 0) out[0] = sigmoid_clipf(red[0]);
}

// ---------------------------------------------------------------------------
// Host-side launcher. setup_inputs() dict order:
//   [0]=side_info, [1]=W0,[2]=N0,[3]=O0,[4]=B0,
//   [5]=W1,[6]=N1,[7]=O1,[8]=B1, [9]=W2,[10]=N2,[11]=O2,[12]=B2
// ---------------------------------------------------------------------------
extern "C" void kernel_launch(void* const* d_in, const int* in_sizes, int n_in,
                              void* d_out, int out_size, void* d_ws, size_t ws_size,
                              hipStream_t stream)
{
    const float* side = (const float*)d_in[0];
    const float* W0 = (const float*)d_in[1];
    const float* N0 = (const float*)d_in[2];
    const float* O0 = (const float*)d_in[3];
    const float* B0 = (const float*)d_in[4];
    const float* W1 = (const float*)d_in[5];
    const float* N1 = (const float*)d_in[6];
    const float* O1 = (const float*)d_in[7];
    const float* B1 = (const float*)d_in[8];
    const float* W2 = (const float*)d_in[9];
    const float* N2 = (const float*)d_in[10];
    const float* O2 = (const float*)d_in[11];
    const float* B2 = (const float*)d_in[12];
    float* out = (float*)d_out;

    // Workspace layout (floats): x0 @ 0 (1025), x1 @ 1088 (1025), x2 @ 2176 (513),
    // gating ints @ float-offset 2752.
    float* ws = (float*)d_ws;
    float* x0 = ws;
    float* x1 = ws + 1088;
    float* x2 = ws + 2176;
    int*   gg = (int*)(ws + 2752);

    gln_setup<<<3, 256, 0, stream>>>(side, N0, N1, N2, O0, O1, O2, B0, B1, B2,
                                     x0, x1, x2, gg);

    // Layer 0: K=1025, N=1024. 4 waves/block * 16 cols = 64 cols/block -> 16 blocks.
    gln_gemv_wmma<<<16, 128, 0, stream>>>(x0, W0, gg, 0, x1 + 1, 1025, 1024, 0);
    // Layer 1: K=1025, N=512 -> 8 blocks.
    gln_gemv_wmma<<<8, 128, 0, stream>>>(x1, W1, gg, 1, x2 + 1, 1025, 512, 0);
    // Layer 2: K=513, N=1 (plain reduction).
    gln_final<<<1, 256, 0, stream>>>(x2, W2, gg, out);
}